// CMDI_10746008175064
// MI455X (gfx1250) — compile-verified
//
#include <hip/hip_runtime.h>

// Native clang vector types (HIP's float4 is a struct; __builtin_nontemporal_*
// needs real vector types to lower to global_{load,store}_b128 th:..NT).
typedef __attribute__((ext_vector_type(4))) float v4f;
typedef __attribute__((ext_vector_type(4))) int   v4i;

// Main fill kernel: out[j] = (flags[j]==1) ? cell[ids[j]] : ctx[j]
// Streamed arrays use non-temporal b128 accesses; the gather uses default
// (regular-temporal) loads so the 8 MB table stays resident in the 192 MB L2.
__global__ __launch_bounds__(256) void cmdi_fill_kernel(
    const float* __restrict__ ctx,
    const int*   __restrict__ flags,
    const int*   __restrict__ ids,
    const float* __restrict__ cell,
    float*       __restrict__ out,
    long long n)
{
    long long i    = (long long)blockIdx.x * blockDim.x + threadIdx.x;
    long long base = i * 4;
    if (base + 3 < n) {
        v4f c = __builtin_nontemporal_load((const v4f*)(ctx   + base)); // b128 NT
        v4i f = __builtin_nontemporal_load((const v4i*)(flags + base)); // b128 NT
        v4i d = __builtin_nontemporal_load((const v4i*)(ids   + base)); // b128 NT
        v4f r;
#pragma unroll
        for (int k = 0; k < 4; ++k) {
            float g = cell[d[k]];                 // RT gather -> L2 resident
            r[k] = (f[k] == 1) ? g : c[k];        // v_cndmask, branch-free
        }
        __builtin_nontemporal_store(r, (v4f*)(out + base));             // b128 NT
    } else if (base < n) {
        // Scalar tail (not reached for the reference sizes; kept for safety).
        for (long long j = base; j < n; ++j) {
            float g = cell[ids[j]];
            out[j] = (flags[j] == 1) ? g : ctx[j];
        }
    }
}

// Second tuple output: copy learning_cell -> out tail. Launched FIRST so its
// regular-temporal loads warm L2 with the whole gather table before the main
// kernel's 51.2M random gathers. Store side is non-temporal (write-once).
__global__ __launch_bounds__(256) void cmdi_copy_cells_kernel(
    const float* __restrict__ cell,
    float*       __restrict__ out_tail,
    long long m)
{
    long long base = ((long long)blockIdx.x * blockDim.x + threadIdx.x) * 4;
    if (base + 3 < m) {
        v4f v = *(const v4f*)(cell + base);                  // RT load: warm L2
        __builtin_nontemporal_store(v, (v4f*)(out_tail + base));
    } else if (base < m) {
        for (long long j = base; j < m; ++j) out_tail[j] = cell[j];
    }
}

extern "C" void kernel_launch(void* const* d_in, const int* in_sizes, int n_in,
                              void* d_out, int out_size, void* d_ws, size_t ws_size,
                              hipStream_t stream) {
    // setup_inputs() order: contexts, learning_cell, missing_flags, cell_ids
    const float* ctx   = (const float*)d_in[0];
    const float* cell  = (const float*)d_in[1];
    const int*   flags = (const int*)  d_in[2];
    const int*   ids   = (const int*)  d_in[3];
    float*       out   = (float*)d_out;

    const long long n = in_sizes[0];   // P*N*W = 51,200,000
    const long long m = in_sizes[1];   // NUM_CELLS = 2,000,000

    // 1) Copy table to output tail (and warm L2 with it): 8 MB, trivial cost.
    {
        long long chunks = (m + 3) / 4;
        int blocks = (int)((chunks + 255) / 256);
        cmdi_copy_cells_kernel<<<blocks, 256, 0, stream>>>(cell, out + n, m);
    }

    // 2) Main masked-gather fill: ~825 MB of NT-streamed traffic + L2 gathers.
    {
        long long chunks = (n + 3) / 4;
        int blocks = (int)((chunks + 255) / 256);
        cmdi_fill_kernel<<<blocks, 256, 0, stream>>>(ctx, flags, ids, cell, out, n);
    }
}